// SparseBoundaryAdd_12438225289334
// MI455X (gfx1250) — compile-verified
//
#include <hip/hip_runtime.h>
#include <hip/hip_bf16.h>

// ---------------------------------------------------------------------------
// SparseBoundaryAdd for MI455X (gfx1250).
//
// map[b,d,i,j] = W[i][j] * (x[b,d,i] + x[b,d,j])
//   W = 0   unmasked, 1 masked off-diagonal, 0.5 diagonal (0.5*2x = x)
//
// Outer sum per 16x16 tile via one V_WMMA_F32_16X16X4_F32:
//   A(16x4) = [ x_rows | ones | 0 | 0 ],  B(4x16) = [ ones^T ; x_cols^T ; 0 ; 0 ]
//   D[i][j] = x_rows[i] + x_cols[j]   (full fp32)
//
// Weights are stored TRANSPOSED (WT[n][m]) so each lane's 8 weights
// (consecutive m, fixed n) are contiguous -> two b128 loads, no serialized
// per-element constant fetches.
// ---------------------------------------------------------------------------

#define N_CLIPS 64

typedef __attribute__((ext_vector_type(2))) float v2f;
typedef __attribute__((ext_vector_type(4))) float v4f;
typedef __attribute__((ext_vector_type(8))) float v8f;

struct WeightTableT {
    float w[N_CLIPS * N_CLIPS];   // transposed: w[n * 64 + m] = W[m][n]
};

static constexpr WeightTableT make_weights_T() {
    WeightTableT t{};
    for (int i = 0; i < N_CLIPS * N_CLIPS; ++i) t.w[i] = 0.0f;
    // diagonal: value must be x[i]; outer sum gives 2*x[i] -> weight 0.5
    for (int i = 0; i < N_CLIPS; ++i) t.w[i * N_CLIPS + i] = 0.5f;
    // strided diagonal offsets: pooling_counts = [15, 8, 8]
    const int counts[3] = {15, 8, 8};
    int stride = 1, offset = 0;
    for (int c = 0; c < 3; ++c) {
        for (int k = 0; k < counts[c]; ++k) {
            offset += stride;
            for (int i = 0; i + offset < N_CLIPS; i += stride) {
                // W[i][i+offset] = 1  ->  WT[(i+offset)*64 + i] = 1
                t.w[(i + offset) * N_CLIPS + i] = 1.0f;
            }
        }
        stride *= 2;
    }
    return t;
}

__constant__ WeightTableT g_weights_T = make_weights_T();

// One block (512 threads = 16 wave32s) per (b,d) slice.
// Wave w computes 16x16 tile (ti = w/4, tj = w%4) of the 64x64 map.
__global__ __launch_bounds__(512) void sba_map_kernel(
    const float* __restrict__ x,   // [16384, 64]
    float* __restrict__ out)       // [16384, 64, 64]
{
    const int slice = blockIdx.x;          // b*D + d
    const int wave  = threadIdx.x >> 5;    // 0..15
    const int lane  = threadIdx.x & 31;    // wave32 lane
    const int ti    = wave >> 2;           // tile row  (0..3)
    const int tj    = wave & 3;            // tile col  (0..3)

    const float* xs = x + (size_t)slice * N_CLIPS;
    float*       o  = out + (size_t)slice * (N_CLIPS * N_CLIPS);

    const int  l16 = lane & 15;
    const bool lo  = lane < 16;            // lanes 0-15 carry K=0/K=1

    // A (16x4 f32): lanes 0-15 -> {K0 = x_rows, K1 = ones}; lanes 16-31 -> 0
    v2f A;
    A.x = lo ? xs[ti * 16 + l16] : 0.0f;
    A.y = lo ? 1.0f              : 0.0f;

    // B (4x16 f32): lanes 0-15 -> {K0 row = ones, K1 row = x_cols}; upper 0
    v2f B;
    B.x = lo ? 1.0f              : 0.0f;
    B.y = lo ? xs[tj * 16 + l16] : 0.0f;

    // C/D layout: VGPR v holds M=v (lanes 0-15) / M=v+8 (lanes 16-31), N=lane%16
    const int n     = tj * 16 + l16;            // global column
    const int mbase = ti * 16 + (lo ? 0 : 8);   // global row base for this half

    // Two wide, independent weight loads (32B aligned: mbase % 8 == 0).
    const float* wrow = g_weights_T.w + n * N_CLIPS + mbase;
    const v4f w0 = *(const v4f*)(wrow);
    const v4f w1 = *(const v4f*)(wrow + 4);

    v8f C = {};
    v8f D = __builtin_amdgcn_wmma_f32_16x16x4_f32(
        /*neg_a=*/false, A, /*neg_b=*/false, B,
        /*c_mod=*/(short)0, C, /*reuse_a=*/false, /*reuse_b=*/false);

    float* op = o + (size_t)mbase * N_CLIPS + n;
#pragma unroll
    for (int v = 0; v < 4; ++v) {
        __builtin_nontemporal_store(D[v] * w0[v], op + (size_t)v * N_CLIPS);
    }
#pragma unroll
    for (int v = 0; v < 4; ++v) {
        __builtin_nontemporal_store(D[v + 4] * w1[v], op + (size_t)(v + 4) * N_CLIPS);
    }
}

// mask2d: [B=32, 1, 64, 64] broadcast of the constant mask (weight != 0).
__global__ __launch_bounds__(256) void sba_mask_kernel(float* __restrict__ mask_out)
{
    const int idx = blockIdx.x * 256 + threadIdx.x;   // 0 .. 32*4096-1
    const int ij  = idx & (N_CLIPS * N_CLIPS - 1);
    const int i   = ij >> 6;
    const int j   = ij & (N_CLIPS - 1);
    // table is transposed: W[i][j] = WT[j*64 + i]
    const float w = g_weights_T.w[j * N_CLIPS + i];
    __builtin_nontemporal_store((w != 0.0f) ? 1.0f : 0.0f, mask_out + idx);
}

extern "C" void kernel_launch(void* const* d_in, const int* in_sizes, int n_in,
                              void* d_out, int out_size, void* d_ws, size_t ws_size,
                              hipStream_t stream)
{
    (void)in_sizes; (void)n_in; (void)d_ws; (void)ws_size; (void)out_size;

    const float* x   = (const float*)d_in[0];        // [32, 512, 64] fp32
    float*       out = (float*)d_out;

    const int B = 32, Dm = 512;
    const int slices = B * Dm;                        // 16384
    const size_t map_elems = (size_t)slices * N_CLIPS * N_CLIPS;  // 67,108,864

    sba_map_kernel<<<slices, 512, 0, stream>>>(x, out);

    float* mask_out = out + map_elems;                // 32*4096 = 131,072 elems
    sba_mask_kernel<<<(B * N_CLIPS * N_CLIPS) / 256, 256, 0, stream>>>(mask_out);
}